// MultiHeadMemory_15874199126502
// MI455X (gfx1250) — compile-verified
//
#include <hip/hip_runtime.h>
#include <stdint.h>

// ---------------------------------------------------------------------------
// MultiHeadMemory (linear-attention associative memory) for MI455X / gfx1250
//   M[b,h] = K^T V  (64x64 f32, reduce over S=4096),  z[b,h] = sum_s k_s
//   out    = (Q M) / (Q.z + eps)
// Bandwidth-bound: 537 MB @ 23.3 TB/s => ~23us floor. f32 WMMA 16x16x4 keeps
// full precision; async-LDS double buffering turns the K/V stream into b128
// DMA traffic overlapped with WMMA.
// ---------------------------------------------------------------------------

typedef __attribute__((ext_vector_type(2))) float v2f;
typedef __attribute__((ext_vector_type(8))) float v8f;
typedef __attribute__((ext_vector_type(4))) unsigned int u32x4;
typedef __attribute__((ext_vector_type(8))) int i32x8;
typedef __attribute__((ext_vector_type(4))) int i32x4;

// async-to-LDS builtin pointer types: param0 = v4i in AS1 (__device__/global),
// param1 = v4i in AS3 (LDS), per round-2 diagnostics.
typedef __attribute__((ext_vector_type(4))) int v4i;
typedef __attribute__((address_space(1))) v4i gas_v4i;
typedef __attribute__((address_space(3))) v4i las_v4i;

#define MHM_EPS 1e-6f
constexpr int Bn = 8, Hn = 16, Sn = 4096, Dn = 64;
constexpr int BH = Bn * Hn;
constexpr int SCHUNK = 256;   // s-values per block in build pass (16 chunks/head)
constexpr int STAGE  = 32;    // s-rows per LDS double-buffer stage
constexpr int KST    = 68;    // padded LDS row stride (floats): 272B, 16B-aligned,
                              // shifts banks by 4/row -> no 2-way half-wave conflicts
constexpr int QCHUNK = 64;    // queries per block in retrieve pass

// ------------------------- feature probes (report via stderr) --------------
#if defined(__has_builtin)
#if __has_builtin(__builtin_amdgcn_tensor_load_to_lds)
#define MHM_HAVE_TDM 1
#endif
#if __has_builtin(__builtin_amdgcn_global_load_async_to_lds_b128)
#define MHM_HAVE_ASYNC 1
#endif
#endif
#ifndef MHM_HAVE_TDM
#define MHM_HAVE_TDM 0
#endif
#ifndef MHM_HAVE_ASYNC
#define MHM_HAVE_ASYNC 0
#endif

#if __has_include(<hip/amd_detail/amd_gfx1250_TDM.h>)
#define MHM_TDM_ARITY6 1
#else
#define MHM_TDM_ARITY6 0
#endif

#if MHM_HAVE_TDM
#pragma message("MHM-PROBE: tensor_load_to_lds builtin PRESENT")
#else
#pragma message("MHM-PROBE: tensor_load_to_lds builtin ABSENT -> manual copy")
#endif
#if MHM_TDM_ARITY6
#pragma message("MHM-PROBE: therock TDM header present -> arity-6 form")
#else
#pragma message("MHM-PROBE: no TDM header -> arity-5 form")
#endif
#if MHM_HAVE_ASYNC
#pragma message("MHM-PROBE: global_load_async_to_lds_b128 builtin PRESENT")
#else
#pragma message("MHM-PROBE: async-load builtin ABSENT -> sync staging")
#endif

#if __has_builtin(__builtin_amdgcn_s_wait_asynccnt)
#pragma message("MHM-PROBE: s_wait_asynccnt builtin PRESENT")
#define MHM_WAIT_ASYNC(n) __builtin_amdgcn_s_wait_asynccnt(n)
#else
#pragma message("MHM-PROBE: s_wait_asynccnt builtin ABSENT -> inline asm")
#define MHM_WAIT_ASYNC(n) asm volatile("s_wait_asynccnt %0" ::"i"(n) : "memory")
#endif

// ---------------------------------------------------------------------------
// Workspace zeroing (harness poisons d_ws once; we must re-zero every launch)
// ---------------------------------------------------------------------------
__global__ void mhm_zero_ws_kernel(float* __restrict__ p, int n) {
  int i = blockIdx.x * blockDim.x + threadIdx.x;
  if (i < n) p[i] = 0.0f;
}

// ---------------------------------------------------------------------------
// Stage STAGE x 64 f32 rows of K and V into padded LDS (stride KST) using
// async b128 copies (ASYNCcnt-tracked) -- 8 async instructions per wave.
// ---------------------------------------------------------------------------
__device__ __forceinline__ void mhm_stage(const float* __restrict__ gK,
                                          const float* __restrict__ gV,
                                          float* ldsK, float* ldsV, int tid) {
#pragma unroll
  for (int i = 0; i < 4; ++i) {
    const int i4 = tid + i * 128;          // float4 index within STAGE*64
    const int r = i4 >> 4;                 // s-row 0..31
    const int c = (i4 & 15) << 2;          // column 0..60 step 4
    const int gi = r * Dn + c;
    const int li = r * KST + c;
#if MHM_HAVE_ASYNC
    __builtin_amdgcn_global_load_async_to_lds_b128(
        (gas_v4i*)(gK + gi), (las_v4i*)(ldsK + li), 0, 0);
    __builtin_amdgcn_global_load_async_to_lds_b128(
        (gas_v4i*)(gV + gi), (las_v4i*)(ldsV + li), 0, 0);
#else
    *(float4*)(ldsK + li) = *(const float4*)(gK + gi);
    *(float4*)(ldsV + li) = *(const float4*)(gV + gi);
#endif
  }
}

// ---------------------------------------------------------------------------
// Pass 1: M[head] += K_chunk^T V_chunk (f32 WMMA 16x16x4), z[head] += sum k
// Block = 128 threads (4 waves). Wave w owns M rows [16w, 16w+16).
// K/V staged in LDS, double-buffered, async-copy overlapped with WMMA.
// ---------------------------------------------------------------------------
__global__ __launch_bounds__(128) void mhm_build_kernel(
    const float* __restrict__ K, const float* __restrict__ V,
    float* __restrict__ Mws, float* __restrict__ zws) {
  __shared__ __align__(16) float Kb[2][STAGE * KST];
  __shared__ __align__(16) float Vb[2][STAGE * KST];

  constexpr int chunks = Sn / SCHUNK;
  const int head = blockIdx.x / chunks;
  const int s0   = (blockIdx.x % chunks) * SCHUNK;
  const int tid  = threadIdx.x;
  const int wave = tid >> 5;
  const int lane = tid & 31;
  const int half = (lane >> 4) & 1;
  const int row  = lane & 15;
  const int mbase = wave * 16;

  const float* __restrict__ Kc = K + ((size_t)head * Sn + s0) * Dn;
  const float* __restrict__ Vc = V + ((size_t)head * Sn + s0) * Dn;

  v8f acc0{}, acc1{}, acc2{}, acc3{};
  float zacc = 0.0f;

  mhm_stage(Kc, Vc, Kb[0], Vb[0], tid);

  constexpr int nst = SCHUNK / STAGE;  // 8 stages
  for (int st = 0; st < nst; ++st) {
    if (st + 1 < nst) {
      // issue next stage into the other buffer, then wait for current stage
      mhm_stage(Kc + (size_t)(st + 1) * STAGE * Dn,
                Vc + (size_t)(st + 1) * STAGE * Dn,
                Kb[(st + 1) & 1], Vb[(st + 1) & 1], tid);
      MHM_WAIT_ASYNC(8);   // 8 outstanding = the stage just issued
    } else {
      MHM_WAIT_ASYNC(0);
    }
    __syncthreads();

    const float* kb = Kb[st & 1];
    const float* vb = Vb[st & 1];
#pragma unroll
    for (int sl = 0; sl < STAGE; sl += 4) {
      const int sa = sl + 2 * half;
      const float ax = kb[sa * KST + mbase + row];
      const float ay = kb[(sa + 1) * KST + mbase + row];
      v2f a;
      a.x = ax;
      a.y = ay;
      zacc += ax + ay;

      const float* v0 = vb + sa * KST + row;
      v2f b0, b1, b2, b3;
      b0.x = v0[0];   b0.y = v0[KST + 0];
      b1.x = v0[16];  b1.y = v0[KST + 16];
      b2.x = v0[32];  b2.y = v0[KST + 32];
      b3.x = v0[48];  b3.y = v0[KST + 48];

      acc0 = __builtin_amdgcn_wmma_f32_16x16x4_f32(false, a, false, b0, (short)0, acc0, false, false);
      acc1 = __builtin_amdgcn_wmma_f32_16x16x4_f32(false, a, false, b1, (short)0, acc1, false, false);
      acc2 = __builtin_amdgcn_wmma_f32_16x16x4_f32(false, a, false, b2, (short)0, acc2, false, false);
      acc3 = __builtin_amdgcn_wmma_f32_16x16x4_f32(false, a, false, b3, (short)0, acc3, false, false);
    }
    __syncthreads();  // buffer free before it is re-filled two stages later
  }

  // C/D layout: VGPR r holds M-row (r + 8*half) across lanes (N = row)
  float* __restrict__ Mh = Mws + (size_t)head * Dn * Dn;
#pragma unroll
  for (int r = 0; r < 8; ++r) {
    const int m = mbase + r + 8 * half;
    atomicAdd(&Mh[(size_t)m * Dn + row +  0], acc0[r]);
    atomicAdd(&Mh[(size_t)m * Dn + row + 16], acc1[r]);
    atomicAdd(&Mh[(size_t)m * Dn + row + 32], acc2[r]);
    atomicAdd(&Mh[(size_t)m * Dn + row + 48], acc3[r]);
  }
  atomicAdd(&zws[(size_t)head * Dn + mbase + row], zacc);
}

// ---------------------------------------------------------------------------
// TDM: async 2D tile load of M (64x64 f32) into LDS via Tensor Data Mover,
// using the TDM LDS-pad feature (4 DWORDs every 64 DWORDs) to write the
// bank-conflict-free stride-KST layout directly.
// ---------------------------------------------------------------------------
#if MHM_HAVE_TDM
__device__ __forceinline__ void mhm_tdm_load_m(const float* gsrc, float* lds_dst) {
  const uint64_t ga = (uint64_t)(uintptr_t)gsrc;
  const uint32_t la = (uint32_t)(uintptr_t)lds_dst;  // flat addr[31:0] == LDS offset

  u32x4 g0;
  g0[0] = 1u;                                                 // count=1 (valid), user mode
  g0[1] = la;                                                 // lds_addr
  g0[2] = (uint32_t)ga;                                       // global_addr[31:0]
  g0[3] = (uint32_t)((ga >> 32) & 0x01FFFFFFu) | (2u << 30);  // global_addr[56:32], type=2

  const uint32_t dim0 = (uint32_t)Dn, dim1 = (uint32_t)Dn;
  i32x8 g1;
  // wg_mask=0 | data_size=2 (4B) | pad_enable | pad_interval=5 (64 DW) | pad_amount=3 (4 DW)
  g1[0] = (int)((2u << 16) | (1u << 20) | (5u << 22) | (3u << 25));
  g1[1] = (int)((dim0 & 0xFFFFu) << 16);                      // tensor_dim0[15:0] @ bits 63:48
  g1[2] = (int)((dim0 >> 16) | ((dim1 & 0xFFFFu) << 16));     // dim0[31:16] | dim1[15:0]
  g1[3] = (int)((dim1 >> 16) | ((dim0 & 0xFFFFu) << 16));     // dim1[31:16] | tile_dim0
  g1[4] = (int)(dim1 & 0xFFFFu);                              // tile_dim1 | tile_dim2=0
  g1[5] = (int)(uint32_t)Dn;                                  // tensor_dim0_stride[31:0]
  g1[6] = 0;
  g1[7] = 0;

  i32x4 g2 = {0, 0, 0, 0};
  i32x4 g3 = {0, 0, 0, 0};
#if MHM_TDM_ARITY6
  i32x8 g4 = {0, 0, 0, 0, 0, 0, 0, 0};
  __builtin_amdgcn_tensor_load_to_lds(g0, g1, g2, g3, g4, 0);
#else
  __builtin_amdgcn_tensor_load_to_lds(g0, g1, g2, g3, 0);
#endif
}
#endif  // MHM_HAVE_TDM

// ---------------------------------------------------------------------------
// Pass 2: out = (Q M) / (Q.z + eps).
// Block = 128 threads (4 waves), 64 queries/block, 16 queries/wave.
// M staged in LDS by TDM (padded); 16 f32-WMMA K-steps per 16x64 tile.
// ---------------------------------------------------------------------------
__global__ __launch_bounds__(128) void mhm_retrieve_kernel(
    const float* __restrict__ Q, const float* __restrict__ Mws,
    const float* __restrict__ zws, float* __restrict__ out) {
  __shared__ __align__(16) float Mlds[Dn * KST];
  __shared__ __align__(16) float zl[Dn];

  constexpr int qchunks = Sn / QCHUNK;
  const int head = blockIdx.x / qchunks;
  const int q0   = (blockIdx.x % qchunks) * QCHUNK;
  const int wave = threadIdx.x >> 5;
  const int lane = threadIdx.x & 31;
  const int half = (lane >> 4) & 1;
  const int row  = lane & 15;

#if MHM_HAVE_TDM
  if (threadIdx.x == 0)
    mhm_tdm_load_m(Mws + (size_t)head * Dn * Dn, Mlds);
  if (threadIdx.x < Dn)
    zl[threadIdx.x] = zws[(size_t)head * Dn + threadIdx.x];
  if (wave == 0)
    __builtin_amdgcn_s_wait_tensorcnt(0);
#else
  {
    const float4* src = (const float4*)(Mws + (size_t)head * Dn * Dn);
#pragma unroll
    for (int i = 0; i < 8; ++i) {
      const int i4 = threadIdx.x + i * 128;   // 1024 float4 total
      const int r = i4 >> 4, c = (i4 & 15) << 2;
      *(float4*)(Mlds + r * KST + c) = src[i4];
    }
    if (threadIdx.x < Dn)
      zl[threadIdx.x] = zws[(size_t)head * Dn + threadIdx.x];
  }
#endif
  __syncthreads();

  const int qb = q0 + wave * 16;
  const float* __restrict__ Qr = Q + ((size_t)head * Sn + qb + row) * Dn;

  // Denominator for query (qb + row): b128 loads, identical in both halves.
  float den = MHM_EPS;
  {
    const float4* q4 = (const float4*)Qr;
#pragma unroll
    for (int d = 0; d < Dn / 4; ++d) {
      const float4 qv = q4[d];
      const float4 zv = *(const float4*)&zl[4 * d];
      den += qv.x * zv.x + qv.y * zv.y + qv.z * zv.z + qv.w * zv.w;
    }
  }
  const float rcp = 1.0f / den;

  v8f acc0{}, acc1{}, acc2{}, acc3{};
#pragma unroll
  for (int d0 = 0; d0 < Dn; d0 += 4) {
    const int da = d0 + 2 * half;
    const v2f a = *(const v2f*)(Qr + da);    // contiguous pair -> b64 load

    const float* m0 = Mlds + da * KST + row;
    v2f b0, b1, b2, b3;
    b0.x = m0[0];   b0.y = m0[KST + 0];
    b1.x = m0[16];  b1.y = m0[KST + 16];
    b2.x = m0[32];  b2.y = m0[KST + 32];
    b3.x = m0[48];  b3.y = m0[KST + 48];

    acc0 = __builtin_amdgcn_wmma_f32_16x16x4_f32(false, a, false, b0, (short)0, acc0, false, false);
    acc1 = __builtin_amdgcn_wmma_f32_16x16x4_f32(false, a, false, b1, (short)0, acc1, false, false);
    acc2 = __builtin_amdgcn_wmma_f32_16x16x4_f32(false, a, false, b2, (short)0, acc2, false, false);
    acc3 = __builtin_amdgcn_wmma_f32_16x16x4_f32(false, a, false, b3, (short)0, acc3, false, false);
  }

  float* __restrict__ Or = out + ((size_t)head * Sn + qb) * Dn;
#pragma unroll
  for (int r = 0; r < 8; ++r) {
    const int m = r + 8 * half;              // output row within the tile
    const float rc = __shfl(rcp, m, 32);     // 1/den for query qb+m (lives in lane m)
    float* orow = Or + (size_t)m * Dn + row;
    orow[0]  = acc0[r] * rc;
    orow[16] = acc1[r] * rc;
    orow[32] = acc2[r] * rc;
    orow[48] = acc3[r] * rc;
  }
}

// ---------------------------------------------------------------------------
// Host launcher
// ---------------------------------------------------------------------------
extern "C" void kernel_launch(void* const* d_in, const int* in_sizes, int n_in,
                              void* d_out, int out_size, void* d_ws, size_t ws_size,
                              hipStream_t stream) {
  (void)in_sizes; (void)n_in; (void)out_size; (void)ws_size;
  const float* keys    = (const float*)d_in[0];
  const float* values  = (const float*)d_in[1];
  const float* queries = (const float*)d_in[2];
  float* out = (float*)d_out;

  float* Mws = (float*)d_ws;                       // BH * 64*64
  float* zws = Mws + (size_t)BH * Dn * Dn;         // BH * 64  (contiguous after M)

  const int nzero = BH * (Dn * Dn + Dn);
  mhm_zero_ws_kernel<<<(nzero + 255) / 256, 256, 0, stream>>>(Mws, nzero);

  mhm_build_kernel<<<BH * (Sn / SCHUNK), 128, 0, stream>>>(keys, values, Mws, zws);

  mhm_retrieve_kernel<<<BH * (Sn / QCHUNK), 128, 0, stream>>>(queries, Mws, zws, out);
}